// GINConv_69638599737428
// MI455X (gfx1250) — compile-verified
//
#include <hip/hip_runtime.h>
#include <cstdint>

// ---------------------------------------------------------------------------
// GINConv (sum aggregator, edge weights):
//   out = (1+eps)*feat + segment_sum(feat[src] * w, dst)
// N=100000 nodes, E=625000 edges, D=128 fp32.
//
// Memory-bound (0.19 GFLOP vs >=110MB HBM). feat (51.2MB) + out (51.2MB) fit
// in the 192MB L2, so gather + atomic scatter run out of L2. CDNA5 path:
//  - TDM gather-mode (tensor_load_to_lds, 32-bit row indices) to DMA feat rows
//    into LDS, double-buffered, overlapped via s_wait_tensorcnt.
//  - wave32 mapping: 1 row (512B) = 32 lanes x float4 (b128 loads).
//  - global_atomic_add_f32 scatter into L2-resident out.
// ---------------------------------------------------------------------------

#define D_FEAT            128
#define ROWS_PER_BATCH    8
#define BATCHES_PER_WAVE  4
#define EDGES_PER_WAVE    (ROWS_PER_BATCH * BATCHES_PER_WAVE)   // 32
#define WAVES_PER_BLOCK   4
#define BLOCK_THREADS     (WAVES_PER_BLOCK * 32)                // 128
#define ROW_FLOATS        D_FEAT                                 // 128
#define BATCH_FLOATS      (ROWS_PER_BATCH * ROW_FLOATS)          // 1024
#define WAVE_LDS_FLOATS   (2 * BATCH_FLOATS)                     // double buffer

typedef unsigned int u32x4 __attribute__((ext_vector_type(4)));
typedef int          i32x8 __attribute__((ext_vector_type(8)));
typedef int          i32x4 __attribute__((ext_vector_type(4)));

// ---------------------------------------------------------------------------
// Kernel 1: out = (1+eps) * feat   (also warms feat & out into L2)
// ---------------------------------------------------------------------------
__global__ void gin_init_kernel(const float* __restrict__ feat,
                                const float* __restrict__ eps,
                                float* __restrict__ out,
                                int n4) {
  int i = blockIdx.x * blockDim.x + threadIdx.x;
  if (i >= n4) return;
  float s = 1.0f + eps[0];                      // uniform -> s_load
  float4 v = reinterpret_cast<const float4*>(feat)[i];   // global_load_b128
  float4 r;
  r.x = v.x * s; r.y = v.y * s; r.z = v.z * s; r.w = v.w * s;
  reinterpret_cast<float4*>(out)[i] = r;                 // global_store_b128
}

// ---------------------------------------------------------------------------
// Kernel 2: TDM-gather + atomic scatter-add. One wave owns 32 edges.
// ---------------------------------------------------------------------------
__global__ __launch_bounds__(BLOCK_THREADS)
void gin_scatter_kernel(const float* __restrict__ feat,
                        const int*   __restrict__ src,
                        const int*   __restrict__ dst,
                        const float* __restrict__ ew,
                        float*       __restrict__ out,
                        int nEdges, int nNodes) {
  __shared__ float lds[WAVES_PER_BLOCK * WAVE_LDS_FLOATS];

  const int lane        = threadIdx.x & 31;
  const int waveInBlock = threadIdx.x >> 5;
  const int waveId      = blockIdx.x * WAVES_PER_BLOCK + waveInBlock;

  // wave-uniform edge range
  const int eBase = __builtin_amdgcn_readfirstlane(waveId * EDGES_PER_WAVE);
  if (eBase >= nEdges) return;
  int edgesHere = nEdges - eBase;
  if (edgesHere > EDGES_PER_WAVE) edgesHere = EDGES_PER_WAVE;
  const int nb = (edgesHere + ROWS_PER_BATCH - 1) / ROWS_PER_BATCH;

  float* myLds = &lds[waveInBlock * WAVE_LDS_FLOATS];
  const uint32_t ldsByte0 = (uint32_t)(uintptr_t)myLds;   // LDS byte offset
  const uint64_t featAddr = (uint64_t)(uintptr_t)feat;

  // ---- issue one TDM gather batch (8 rows of 512B) into buffer (b&1) ----
  auto issue = [&](int b) {
    const int e0  = eBase + b * ROWS_PER_BATCH;
    int cnt = edgesHere - b * ROWS_PER_BATCH;
    if (cnt > ROWS_PER_BATCH) cnt = ROWS_PER_BATCH;

    int idx[ROWS_PER_BATCH];
#pragma unroll
    for (int i = 0; i < ROWS_PER_BATCH; ++i)
      idx[i] = (i < cnt) ? src[e0 + i] : 0;               // uniform -> s_load

    // D# group 0: count=1, gather_mode=1, 32-bit indices, type=2 ("image")
    u32x4 g0;
    g0[0] = 0xC0000001u;                                  // count=1|gidx32|gather
    g0[1] = ldsByte0 + (uint32_t)(b & 1) * (BATCH_FLOATS * 4u);
    g0[2] = (uint32_t)featAddr;
    g0[3] = ((uint32_t)(featAddr >> 32) & 0x01FFFFFFu) | 0x80000000u;

    // D# group 1: data_size=4B, tensor_dim0=128, tensor_dim1=nNodes,
    //             tile_dim0=128, tile_dim1=cnt, dim0_stride=128
    i32x8 g1;
    g1[0] = 0x00020000;                                   // data_size=2 (4B)
    g1[1] = (int)((uint32_t)D_FEAT << 16);                // tensor_dim0 lo16
    g1[2] = (int)(((uint32_t)nNodes & 0xFFFFu) << 16);    // dim0 hi16=0 | dim1 lo16
    g1[3] = (int)(((uint32_t)nNodes >> 16) |
                  ((uint32_t)D_FEAT << 16));              // dim1 hi16 | tile_dim0
    g1[4] = cnt;                                          // tile_dim1 = #indices
    g1[5] = D_FEAT;                                       // dim0_stride lo32
    g1[6] = 0;
    g1[7] = 0;

    // D# groups 2/3: 8 x 32-bit row indices
    i32x4 g2, g3;
    g2[0] = idx[0]; g2[1] = idx[1]; g2[2] = idx[2]; g2[3] = idx[3];
    g3[0] = idx[4]; g3[1] = idx[5]; g3[2] = idx[6]; g3[3] = idx[7];

#if defined(__clang_major__) && (__clang_major__ >= 23)
    i32x8 gpad = {};
    __builtin_amdgcn_tensor_load_to_lds(g0, g1, g2, g3, gpad, 0);
#else
    __builtin_amdgcn_tensor_load_to_lds(g0, g1, g2, g3, 0);
#endif
  };

  // ---- consume one batch: scale by edge weight, atomic-add into out ----
  auto process = [&](int b) {
    const float4* lbuf = reinterpret_cast<const float4*>(
        myLds + (b & 1) * BATCH_FLOATS);
    const int e0  = eBase + b * ROWS_PER_BATCH;
    int cnt = edgesHere - b * ROWS_PER_BATCH;
    if (cnt > ROWS_PER_BATCH) cnt = ROWS_PER_BATCH;

    for (int i = 0; i < cnt; ++i) {
      const int   e = e0 + i;
      const int   d = __builtin_amdgcn_readfirstlane(dst[e]); // SGPR base
      const float w = ew[e];                                  // uniform
      float4 v = lbuf[i * (ROW_FLOATS / 4) + lane];           // ds_load_b128
      float* o = out + (size_t)d * D_FEAT + lane * 4;
      (void)__hip_atomic_fetch_add(o + 0, v.x * w, __ATOMIC_RELAXED,
                                   __HIP_MEMORY_SCOPE_AGENT);
      (void)__hip_atomic_fetch_add(o + 1, v.y * w, __ATOMIC_RELAXED,
                                   __HIP_MEMORY_SCOPE_AGENT);
      (void)__hip_atomic_fetch_add(o + 2, v.z * w, __ATOMIC_RELAXED,
                                   __HIP_MEMORY_SCOPE_AGENT);
      (void)__hip_atomic_fetch_add(o + 3, v.w * w, __ATOMIC_RELAXED,
                                   __HIP_MEMORY_SCOPE_AGENT);
    }
  };

  // ---- double-buffered pipeline: overlap DMA of batch b+1 with batch b ----
  issue(0);
  for (int b = 0; b + 1 < nb; ++b) {
    issue(b + 1);
    __builtin_amdgcn_s_wait_tensorcnt(1);   // oldest DMA done
    process(b);
  }
  __builtin_amdgcn_s_wait_tensorcnt(0);
  process(nb - 1);
}

// ---------------------------------------------------------------------------
extern "C" void kernel_launch(void* const* d_in, const int* in_sizes, int n_in,
                              void* d_out, int out_size, void* d_ws, size_t ws_size,
                              hipStream_t stream) {
  const float* feat = (const float*)d_in[0];
  const int*   src  = (const int*)d_in[1];
  const int*   dst  = (const int*)d_in[2];
  const float* ew   = (const float*)d_in[3];
  const float* eps  = (const float*)d_in[4];
  float*       out  = (float*)d_out;

  const int nNodes = in_sizes[0] / D_FEAT;
  const int nEdges = in_sizes[1];
  const int n4     = in_sizes[0] / 4;

  gin_init_kernel<<<(n4 + 255) / 256, 256, 0, stream>>>(feat, eps, out, n4);

  const int waves  = (nEdges + EDGES_PER_WAVE - 1) / EDGES_PER_WAVE;
  const int blocks = (waves + WAVES_PER_BLOCK - 1) / WAVES_PER_BLOCK;
  gin_scatter_kernel<<<blocks, BLOCK_THREADS, 0, stream>>>(
      feat, src, dst, ew, out, nEdges, nNodes);
}